// GCNClassifier_36979668418700
// MI455X (gfx1250) — compile-verified
//
#include <hip/hip_runtime.h>

// Problem dimensions (fixed by the reference).
constexpr int NN = 20000;   // nodes
constexpr int EE = 320000;  // edges
constexpr int FF = 64;      // input features
constexpr int HH = 256;     // hidden
constexpr int CC = 10;      // classes
constexpr int GG = 64;      // graphs

typedef __attribute__((ext_vector_type(2))) float v2f;
typedef __attribute__((ext_vector_type(8))) float v8f;

// ================= CSR construction =================
__global__ void zero_int(int* __restrict__ counts_i, int* __restrict__ cursor,
                         int* __restrict__ gcount) {
    int i = blockIdx.x * blockDim.x + threadIdx.x;
    if (i < NN) { counts_i[i] = 0; cursor[i] = 0; }
    if (i < GG) gcount[i] = 0;
}

__global__ void count_indeg(const int* __restrict__ col, int* __restrict__ counts_i) {
    int e = blockIdx.x * blockDim.x + threadIdx.x;
    if (e < EE) atomicAdd(&counts_i[col[e]], 1);
}

// Single-workgroup exclusive scan: 1024 threads x 20 elements.
__global__ void __launch_bounds__(1024)
scan_offsets(const int* __restrict__ counts_i, int* __restrict__ offsets) {
    __shared__ int part[1024];
    const int t = threadIdx.x;
    const int CH = (NN + 1023) / 1024;   // 20
    const int base = t * CH;
    int s = 0;
    for (int j = 0; j < CH; ++j) {
        int idx = base + j;
        if (idx < NN) s += counts_i[idx];
    }
    part[t] = s;
    __syncthreads();
    for (int off = 1; off < 1024; off <<= 1) {
        int v = (t >= off) ? part[t - off] : 0;
        __syncthreads();
        part[t] += v;
        __syncthreads();
    }
    int run = (t == 0) ? 0 : part[t - 1];
    for (int j = 0; j < CH; ++j) {
        int idx = base + j;
        if (idx < NN) { offsets[idx] = run; run += counts_i[idx]; }
    }
    if (t == 1023) offsets[NN] = part[1023];   // == EE
}

__global__ void csr_fill(const int* __restrict__ row, const int* __restrict__ col,
                         const int* __restrict__ offsets, int* __restrict__ cursor,
                         int* __restrict__ adj) {
    int e = blockIdx.x * blockDim.x + threadIdx.x;
    if (e >= EE) return;
    int c = col[e];
    int p = atomicAdd(&cursor[c], 1);
    adj[offsets[c] + p] = row[e];
}

// Make adjacency lists a deterministic function of the input (fill order via
// atomics is run-dependent): per-node insertion sort, avg degree ~16.
__global__ void sort_adj(const int* __restrict__ offsets, int* __restrict__ adj) {
    int v = blockIdx.x * blockDim.x + threadIdx.x;
    if (v >= NN) return;
    int beg = offsets[v], end = offsets[v + 1];
    for (int i = beg + 1; i < end; ++i) {
        int key = adj[i];
        int j = i - 1;
        while (j >= beg && adj[j] > key) { adj[j + 1] = adj[j]; --j; }
        adj[j + 1] = key;
    }
}

__global__ void dinv_kernel(const int* __restrict__ counts_i, float* __restrict__ dinv) {
    int i = blockIdx.x * blockDim.x + threadIdx.x;
    if (i < NN) dinv[i] = rsqrtf(1.0f + (float)counts_i[i]);   // self-loop adds 1
}

// ================= fp32 WMMA GEMM: Y[N x H] = X[N x K] * W[K x H] =================
// One wave computes a 32x32 tile (2x2 sub-tiles of 16x16); 4 waves/block span 128 cols.
// A layout (16x4 f32): lane l holds row (l&15), K = (l>>4)*2 + {0,1}
// B layout (4x16 f32): lane l holds col (l&15), K = (l>>4)*2 + {0,1}
// D layout: acc[i] = D[8*(l>>4) + i][l&15]
__global__ void __launch_bounds__(128)
gemm_wmma_f32(const float* __restrict__ X, const float* __restrict__ Wm,
              float* __restrict__ Y, int K) {
    const int lane = threadIdx.x & 31;
    const int wave = threadIdx.x >> 5;
    const int m0 = blockIdx.x * 32;
    const int n0 = blockIdx.y * 128 + wave * 32;

    const int lrow  = lane & 15;
    const int khalf = (lane >> 4) * 2;   // 0 or 2

    const float* __restrict__ arow0 = X + (size_t)(m0 + lrow) * K;
    const float* __restrict__ arow1 = arow0 + (size_t)16 * K;
    const float* __restrict__ bcol0 = Wm + (n0 + lrow);
    const float* __restrict__ bcol1 = bcol0 + 16;

    v8f acc00 = {}, acc01 = {}, acc10 = {}, acc11 = {};
    for (int k = 0; k < K; k += 4) {
        const int kk = k + khalf;
        v2f a0 = *(const v2f*)(arow0 + kk);
        v2f a1 = *(const v2f*)(arow1 + kk);
        v2f b0, b1;
        b0.x = bcol0[(size_t)(kk + 0) * HH];
        b0.y = bcol0[(size_t)(kk + 1) * HH];
        b1.x = bcol1[(size_t)(kk + 0) * HH];
        b1.y = bcol1[(size_t)(kk + 1) * HH];
        acc00 = __builtin_amdgcn_wmma_f32_16x16x4_f32(false, a0, false, b0, (short)0, acc00, false, false);
        acc01 = __builtin_amdgcn_wmma_f32_16x16x4_f32(false, a0, false, b1, (short)0, acc01, false, false);
        acc10 = __builtin_amdgcn_wmma_f32_16x16x4_f32(false, a1, false, b0, (short)0, acc10, false, false);
        acc11 = __builtin_amdgcn_wmma_f32_16x16x4_f32(false, a1, false, b1, (short)0, acc11, false, false);
    }

    const int mbase = m0 + 8 * (lane >> 4);
    float* __restrict__ d00 = Y + (size_t)mbase * HH + (n0 + lrow);
#pragma unroll
    for (int i = 0; i < 8; ++i) {
        d00[(size_t)i * HH]              = acc00[i];
        d00[(size_t)i * HH + 16]         = acc01[i];
        d00[(size_t)(i + 16) * HH]       = acc10[i];
        d00[(size_t)(i + 16) * HH + 16]  = acc11[i];
    }
}

// ================= fused aggregation: Hout[v] = relu(sum + bias) =================
// One wave per destination node; each lane owns 8 of the 256 columns.
// sum = dinv[v]^2 * Y[v]  +  sum_{s in adj(v)} dinv[v]*dinv[s] * Y[s]
__global__ void __launch_bounds__(256)
aggregate(const float* __restrict__ Y, const int* __restrict__ offsets,
          const int* __restrict__ adj, const float* __restrict__ dinv,
          const float* __restrict__ bias, float* __restrict__ Hout) {
    int v = blockIdx.x * (blockDim.x >> 5) + (threadIdx.x >> 5);
    v = __builtin_amdgcn_readfirstlane(v);        // wave-uniform -> scalar loads
    if (v >= NN) return;
    const int lane = threadIdx.x & 31;

    const float dv = dinv[v];
    const float4* __restrict__ srcv = (const float4*)(Y + (size_t)v * HH);
    float4 a = srcv[lane];
    float4 b = srcv[lane + 32];
    float ws = dv * dv;
    float4 acc0, acc1;
    acc0.x = ws * a.x; acc0.y = ws * a.y; acc0.z = ws * a.z; acc0.w = ws * a.w;
    acc1.x = ws * b.x; acc1.y = ws * b.y; acc1.z = ws * b.z; acc1.w = ws * b.w;

    const int beg = offsets[v];
    const int end = offsets[v + 1];
    for (int i = beg; i < end; ++i) {
        int s = adj[i];
        if (i + 1 < end)
            __builtin_prefetch(Y + (size_t)adj[i + 1] * HH, 0, 1);  // global_prefetch
        float w = dv * dinv[s];
        const float4* __restrict__ srcs = (const float4*)(Y + (size_t)s * HH);
        float4 sa = srcs[lane];
        float4 sb = srcs[lane + 32];
        acc0.x = fmaf(w, sa.x, acc0.x); acc0.y = fmaf(w, sa.y, acc0.y);
        acc0.z = fmaf(w, sa.z, acc0.z); acc0.w = fmaf(w, sa.w, acc0.w);
        acc1.x = fmaf(w, sb.x, acc1.x); acc1.y = fmaf(w, sb.y, acc1.y);
        acc1.z = fmaf(w, sb.z, acc1.z); acc1.w = fmaf(w, sb.w, acc1.w);
    }

    const int c0 = lane * 4;
    float4 bb0 = *(const float4*)(bias + c0);
    float4 bb1 = *(const float4*)(bias + 128 + c0);
    float4 o0, o1;
    o0.x = fmaxf(acc0.x + bb0.x, 0.0f); o0.y = fmaxf(acc0.y + bb0.y, 0.0f);
    o0.z = fmaxf(acc0.z + bb0.z, 0.0f); o0.w = fmaxf(acc0.w + bb0.w, 0.0f);
    o1.x = fmaxf(acc1.x + bb1.x, 0.0f); o1.y = fmaxf(acc1.y + bb1.y, 0.0f);
    o1.z = fmaxf(acc1.z + bb1.z, 0.0f); o1.w = fmaxf(acc1.w + bb1.w, 0.0f);

    float4* __restrict__ dst = (float4*)(Hout + (size_t)v * HH);
    dst[lane]      = o0;
    dst[lane + 32] = o1;
}

// ================= deterministic pooling (batch is sorted) =================
__global__ void gcount_kernel(const int* __restrict__ batch, int* __restrict__ gcount) {
    int i = blockIdx.x * blockDim.x + threadIdx.x;
    if (i < NN) atomicAdd(&gcount[batch[i]], 1);   // int atomics: order-independent
}

__global__ void gscan_kernel(const int* __restrict__ gcount, int* __restrict__ gstart) {
    if (threadIdx.x == 0 && blockIdx.x == 0) {
        int run = 0;
        for (int g = 0; g < GG; ++g) { gstart[g] = run; run += gcount[g]; }
    }
}

// One block per graph; thread t owns column t; sequential (deterministic) sum
// over the graph's contiguous node range, then mean.
__global__ void __launch_bounds__(256)
pool_mean(const float* __restrict__ Hh, const int* __restrict__ gstart,
          const int* __restrict__ gcount, float* __restrict__ pooled) {
    int g = blockIdx.x;
    int t = threadIdx.x;
    int s = gstart[g];
    int n = gcount[g];
    float acc = 0.0f;
    for (int i = 0; i < n; ++i)
        acc += Hh[(size_t)(s + i) * HH + t];
    pooled[(size_t)g * HH + t] = acc / fmaxf((float)n, 1.0f);
}

// ================= classifier head: out = pooled_mean @ Wout + bout =================
__global__ void head_kernel(const float* __restrict__ pooled,
                            const float* __restrict__ Wout,
                            const float* __restrict__ bout,
                            float* __restrict__ out) {
    int i = blockIdx.x * blockDim.x + threadIdx.x;
    if (i >= GG * CC) return;
    int g = i / CC, c = i % CC;
    const float* p = pooled + (size_t)g * HH;
    float s = 0.0f;
    for (int h = 0; h < HH; ++h) s += p[h] * Wout[(size_t)h * CC + c];
    out[i] = s + bout[c];
}

// ================= host launcher =================
extern "C" void kernel_launch(void* const* d_in, const int* in_sizes, int n_in,
                              void* d_out, int out_size, void* d_ws, size_t ws_size,
                              hipStream_t stream) {
    const float* x     = (const float*)d_in[0];
    const int*   ei    = (const int*)  d_in[1];   // [2, E]
    const int*   batch = (const int*)  d_in[2];
    const float* W1 = (const float*)d_in[3];  const float* b1 = (const float*)d_in[4];
    const float* W2 = (const float*)d_in[5];  const float* b2 = (const float*)d_in[6];
    const float* W3 = (const float*)d_in[7];  const float* b3 = (const float*)d_in[8];
    const float* W4 = (const float*)d_in[9];  const float* b4 = (const float*)d_in[10];
    const float* W5 = (const float*)d_in[11]; const float* b5 = (const float*)d_in[12];
    const float* Wout = (const float*)d_in[13];
    const float* bout = (const float*)d_in[14];
    float* out = (float*)d_out;

    const int* row = ei;        // edge_index[0] (sources)
    const int* col = ei + EE;   // edge_index[1] (targets)

    float* ws     = (float*)d_ws;
    float* dinv   = ws;                            // N
    float* Y      = dinv + NN;                     // N*H
    float* Ha     = Y  + (size_t)NN * HH;          // N*H
    float* Hb     = Ha + (size_t)NN * HH;          // N*H
    float* pooled = Hb + (size_t)NN * HH;          // G*H (mean)
    int* counts_i = (int*)(pooled + (size_t)GG * HH); // N
    int* offsets  = counts_i + NN;                 // N+1
    int* cursor   = offsets + NN + 1;              // N
    int* adj      = cursor + NN;                   // E
    int* gcount   = adj + EE;                      // G
    int* gstart   = gcount + GG;                   // G

    // ---- CSR + normalization + graph segments (once) ----
    zero_int     <<<(NN + 255) / 256, 256, 0, stream>>>(counts_i, cursor, gcount);
    count_indeg  <<<(EE + 255) / 256, 256, 0, stream>>>(col, counts_i);
    scan_offsets <<<1, 1024, 0, stream>>>(counts_i, offsets);
    csr_fill     <<<(EE + 255) / 256, 256, 0, stream>>>(row, col, offsets, cursor, adj);
    sort_adj     <<<(NN + 255) / 256, 256, 0, stream>>>(offsets, adj);
    dinv_kernel  <<<(NN + 255) / 256, 256, 0, stream>>>(counts_i, dinv);
    gcount_kernel<<<(NN + 255) / 256, 256, 0, stream>>>(batch, gcount);
    gscan_kernel <<<1, 64, 0, stream>>>(gcount, gstart);

    const dim3 gemmGrid(NN / 32, HH / 128);   // 625 x 2
    const int aggBlocks = (NN + 7) / 8;       // 8 waves/block, 1 node/wave

    const float* Hin = x;
    int K = FF;
    float* bufs[2] = { Ha, Hb };
    const float* Ws[5] = { W1, W2, W3, W4, W5 };
    const float* bs[5] = { b1, b2, b3, b4, b5 };

    for (int l = 0; l < 5; ++l) {
        float* Hout = bufs[l & 1];
        gemm_wmma_f32<<<gemmGrid, 128, 0, stream>>>(Hin, Ws[l], Y, K);
        aggregate<<<aggBlocks, 256, 0, stream>>>(Y, offsets, adj, dinv, bs[l], Hout);
        Hin = Hout;
        K = HH;
    }

    // ---- deterministic global mean pool + head ----
    pool_mean  <<<GG, 256, 0, stream>>>(Hin, gstart, gcount, pooled);
    head_kernel<<<(GG * CC + 255) / 256, 256, 0, stream>>>(pooled, Wout, bout, out);
}